// GCN_model_22694607192213
// MI455X (gfx1250) — compile-verified
//
#include <hip/hip_runtime.h>
#include <hip/hip_bf16.h>

typedef float v2f __attribute__((ext_vector_type(2)));
typedef float v8f __attribute__((ext_vector_type(8)));

#define G_CT 20
#define HDIM 64
#define KCL  16

// ---------------------------------------------------------------- utilities
__global__ void zero_f32(float* __restrict__ p, long long n) {
  long long i = (long long)blockIdx.x * blockDim.x + threadIdx.x;
  if (i < n) p[i] = 0.0f;
}

// deg[dst] += 1 per edge  (segment_sum(ones, dst))
__global__ void deg_count(const int* __restrict__ ei, float* __restrict__ deg, int E) {
  int e = blockIdx.x * blockDim.x + threadIdx.x;
  if (e < E) atomicAdd(&deg[ei[E + e]], 1.0f);
}

__global__ void compute_dis(const float* __restrict__ deg, float* __restrict__ dis, int n) {
  int i = blockIdx.x * blockDim.x + threadIdx.x;
  if (i < n) dis[i] = rsqrtf(deg[i] + 1.0f);
}

// ----------------------------------------------- h0 = relu([onehot|feats] @ W0 + b0)
__global__ void embed_gemm0(const float* __restrict__ x, const float* __restrict__ W0,
                            const float* __restrict__ b0, float* __restrict__ h0, int n) {
  int t = blockIdx.x * blockDim.x + threadIdx.x;
  if (t >= n * HDIM) return;
  int i = t >> 6, j = t & 63;
  const float* xr = x + (size_t)i * 41;
  int ct = (int)xr[0];                       // one-hot row select
  float acc = b0[j] + W0[ct * HDIM + j];
#pragma unroll
  for (int k = 0; k < 40; ++k) acc += xr[1 + k] * W0[(G_CT + k) * HDIM + j];
  h0[t] = fmaxf(acc, 0.0f);
}

// ----------------------------------------------- out(n x 64) = A(n x 64) @ W(64 x 64)
// One wave per 16x16 output tile, V_WMMA_F32_16X16X4_F32 chained over K=64.
// Weight tile staged global->LDS via GLOBAL_LOAD_ASYNC_TO_LDS_B128 (ASYNCcnt path).
__global__ __launch_bounds__(256) void gemm_h_w_wmma(const float* __restrict__ A,
                                                     const float* __restrict__ W,
                                                     float* __restrict__ out, int n) {
  __shared__ __align__(16) float sW[HDIM * HDIM];   // 16 KB weight tile

#if defined(__AMDGCN__)
  {
    // Generic LDS pointers carry the LDS byte offset in their low 32 bits.
    unsigned ldsoff = (unsigned)(uintptr_t)(void*)&sW[0] + threadIdx.x * 16u;
    const float* gsrc = W + threadIdx.x * 4;        // 16 bytes per lane per round
#pragma unroll
    for (int r = 0; r < 4; ++r) {                   // 4 x 4KB = 16KB
      asm volatile("global_load_async_to_lds_b128 %0, %1, off"
                   :: "v"(ldsoff + (unsigned)(r * 4096)),
                      "v"((const void*)(gsrc + r * 1024))
                   : "memory");
    }
    asm volatile("s_wait_asynccnt 0x0" ::: "memory");
  }
  __syncthreads();
#else
  for (int t = threadIdx.x; t < HDIM * HDIM; t += 256) sW[t] = W[t];
  __syncthreads();
#endif

  int wave  = threadIdx.x >> 5;
  int lane  = threadIdx.x & 31;
  int ctile = wave & 3;                       // 4 column tiles of 16
  int rtile = blockIdx.x * 2 + (wave >> 2);   // 2 row tiles per block
  int m0 = rtile * 16;
  if (m0 >= n) return;                        // wave-uniform: EXEC stays all-1s

  int lr = lane & 15;                         // row (A) / col (B,D) within tile
  int hi = lane >> 4;                         // lane-half selects K pairs
  const float* arow = A + (size_t)(m0 + lr) * HDIM;
  const float* bcol = sW + ctile * 16 + lr;

  v8f c = {};
#pragma unroll
  for (int kk = 0; kk < 16; ++kk) {
    int k0 = kk * 4 + hi * 2;
    v2f a, b;
    // A 16x4 f32 layout: v0 = K{0|2}, v1 = K{1|3} across lane halves
    a.x = arow[k0];            a.y = arow[k0 + 1];
    // B 4x16 f32 layout: v0 = row K{0|2}, v1 = row K{1|3}, N = lane&15
    b.x = bcol[(size_t)k0 * HDIM];
    b.y = bcol[(size_t)(k0 + 1) * HDIM];
    c = __builtin_amdgcn_wmma_f32_16x16x4_f32(false, a, false, b, (short)0, c,
                                              false, false);
  }
  // D 16x16 f32 layout: VGPR v -> M = v + hi*8, N = lane&15
#pragma unroll
  for (int v = 0; v < 8; ++v)
    out[(size_t)(m0 + v + hi * 8) * HDIM + ctile * 16 + lr] = c[v];
}

// ----------------------------------------------- agg[dst] += hW[src] * dis[src]*dis[dst]
__global__ void edge_scatter(const int* __restrict__ ei, const float* __restrict__ dis,
                             const float* __restrict__ hW, float* __restrict__ agg, int E) {
  long long gid = (long long)blockIdx.x * blockDim.x + threadIdx.x;
  int e = (int)(gid >> 4);
  if (e >= E) return;
  int q = ((int)gid & 15) << 2;               // 16 threads cover the 64-float row
  int src = ei[e], dst = ei[E + e];
  float nrm = dis[src] * dis[dst];
  const float4 v = *reinterpret_cast<const float4*>(hW + (size_t)src * HDIM + q);
  float* p = agg + (size_t)dst * HDIM + q;
  atomicAdd(p + 0, v.x * nrm);
  atomicAdd(p + 1, v.y * nrm);
  atomicAdd(p + 2, v.z * nrm);
  atomicAdd(p + 3, v.w * nrm);
}

// ----------------------------------------------- agg += hW*dis^2 + b (+res), relu
__global__ void conv_finalize(float* __restrict__ agg, const float* __restrict__ hW,
                              const float* __restrict__ dis, const float* __restrict__ b,
                              const float* __restrict__ res, int do_relu, int n) {
  int t = blockIdx.x * blockDim.x + threadIdx.x;
  if (t >= n * HDIM) return;
  int i = t >> 6, j = t & 63;
  float d = dis[i];
  float v = agg[t] + hW[t] * d * d + b[j];
  if (res) v += res[t];
  if (do_relu) v = fmaxf(v, 0.0f);
  agg[t] = v;
}

// ----------------------------------------------- s = softmax(out @ Wl + bl)
__global__ void cluster_softmax(const float* __restrict__ h, const float* __restrict__ Wl,
                                const float* __restrict__ bl, float* __restrict__ s, int n) {
  int i = blockIdx.x * blockDim.x + threadIdx.x;
  if (i >= n) return;
  float acc[KCL];
#pragma unroll
  for (int c = 0; c < KCL; ++c) acc[c] = bl[c];
  const float* hr = h + (size_t)i * HDIM;
  for (int k = 0; k < HDIM; ++k) {
    float xv = hr[k];
    const float* wr = Wl + k * KCL;
#pragma unroll
    for (int c = 0; c < KCL; ++c) acc[c] += xv * wr[c];
  }
  float mx = acc[0];
#pragma unroll
  for (int c = 1; c < KCL; ++c) mx = fmaxf(mx, acc[c]);
  float sum = 0.f;
#pragma unroll
  for (int c = 0; c < KCL; ++c) { acc[c] = __expf(acc[c] - mx); sum += acc[c]; }
  float inv = 1.0f / sum;
  float* sr = s + (size_t)i * KCL;
#pragma unroll
  for (int c = 0; c < KCL; ++c) sr[c] = acc[c] * inv;
}

// ----------------------------------------------- num = sum_e dot(s[src], s[dst])
__global__ __launch_bounds__(256) void edge_num_reduce(const int* __restrict__ ei,
                                                       const float* __restrict__ s,
                                                       float* __restrict__ red, int E) {
  __shared__ float sh[256];
  int e = blockIdx.x * blockDim.x + threadIdx.x;
  float v = 0.f;
  if (e < E) {
    const float* a = s + (size_t)ei[e] * KCL;
    const float* b = s + (size_t)ei[E + e] * KCL;
#pragma unroll
    for (int c = 0; c < KCL; ++c) v += a[c] * b[c];
  }
  sh[threadIdx.x] = v;
  __syncthreads();
  for (int off = 128; off > 0; off >>= 1) {
    if (threadIdx.x < off) sh[threadIdx.x] += sh[threadIdx.x + off];
    __syncthreads();
  }
  if (threadIdx.x == 0) atomicAdd(&red[0], sh[0]);
}

// ------------------------------- den = sum_i deg_i*||s_i||^2 ; ss += s_blk^T s_blk
__global__ __launch_bounds__(256) void den_ss_reduce(const float* __restrict__ s,
                                                     const float* __restrict__ deg,
                                                     float* __restrict__ red,
                                                     float* __restrict__ ss, int n) {
  __shared__ float sh[256 * KCL];
  __shared__ float rsh[256];
  int tid = threadIdx.x;
  int i = blockIdx.x * 256 + tid;
  float dval = 0.f;
  if (i < n) {
    float ssq = 0.f;
#pragma unroll
    for (int c = 0; c < KCL; ++c) {
      float v = s[(size_t)i * KCL + c];
      sh[tid * KCL + c] = v;
      ssq += v * v;
    }
    dval = deg[i] * ssq;
  } else {
#pragma unroll
    for (int c = 0; c < KCL; ++c) sh[tid * KCL + c] = 0.f;
  }
  rsh[tid] = dval;
  __syncthreads();
  // each thread owns one entry of the 16x16 Gram matrix
  int ra = tid >> 4, rb = tid & 15;
  float acc = 0.f;
#pragma unroll 8
  for (int nn = 0; nn < 256; ++nn) acc += sh[nn * KCL + ra] * sh[nn * KCL + rb];
  atomicAdd(&ss[tid], acc);
  for (int off = 128; off > 0; off >>= 1) {
    if (tid < off) rsh[tid] += rsh[tid + off];
    __syncthreads();
  }
  if (tid == 0) atomicAdd(&red[1], rsh[0]);
}

// ----------------------------------------------- final scalars: mc, ortho
__global__ __launch_bounds__(256) void final_losses(const float* __restrict__ red,
                                                    const float* __restrict__ ss,
                                                    float* __restrict__ out) {
  __shared__ float sh[256];
  int t = threadIdx.x;
  float v = ss[t];
  sh[t] = v * v;
  __syncthreads();
  for (int off = 128; off > 0; off >>= 1) {
    if (t < off) sh[t] += sh[t + off];
    __syncthreads();
  }
  float fro = sqrtf(sh[0]);
  __syncthreads();
  float eye = ((t >> 4) == (t & 15)) ? 0.25f : 0.0f;  // I/sqrt(16)
  float d = v / fro - eye;
  sh[t] = d * d;
  __syncthreads();
  for (int off = 128; off > 0; off >>= 1) {
    if (t < off) sh[t] += sh[t + off];
    __syncthreads();
  }
  if (t == 0) {
    out[0] = -(red[0] / red[1]);   // mincut loss
    out[1] = sqrtf(sh[0]);         // ortho loss
  }
}

// ======================================================================== launch
static inline int cdiv_ll(long long a, long long b) { return (int)((a + b - 1) / b); }

extern "C" void kernel_launch(void* const* d_in, const int* in_sizes, int n_in,
                              void* d_out, int out_size, void* d_ws, size_t ws_size,
                              hipStream_t stream) {
  const float* x  = (const float*)d_in[0];
  const int*   ei = (const int*)d_in[1];
  // d_in[2]=batch, d_in[3]=n_cells (unused; n from batch length)
  const float* W0 = (const float*)d_in[4];
  const float* b0 = (const float*)d_in[5];
  const float* W1 = (const float*)d_in[6];
  const float* b1 = (const float*)d_in[7];
  const float* W2 = (const float*)d_in[8];
  const float* b2 = (const float*)d_in[9];
  const float* W3 = (const float*)d_in[10];
  const float* b3 = (const float*)d_in[11];
  const float* Wl = (const float*)d_in[12];
  const float* bl = (const float*)d_in[13];

  const int n = in_sizes[2];        // 50000
  const int E = in_sizes[1] / 2;    // 800000
  const long long nh = (long long)n * HDIM;

  float* ws = (float*)d_ws;
  size_t off = 0;
  float* deg  = ws + off; off += (size_t)n;
  float* dis  = ws + off; off += (size_t)n;
  float* h0   = ws + off; off += (size_t)n * HDIM;
  float* tmp  = ws + off; off += (size_t)n * HDIM;
  float* buf1 = ws + off; off += (size_t)n * HDIM;
  float* buf2 = ws + off; off += (size_t)n * HDIM;
  float* sm   = ws + off; off += (size_t)n * KCL;
  float* ssm  = ws + off; off += 256;
  float* red  = ws + off; off += 8;

  const int T = 256;
  const int gemmGrid = cdiv_ll(cdiv_ll(n, 16), 2);

  // degrees + normalization
  zero_f32   <<<cdiv_ll(n, T), T, 0, stream>>>(deg, n);
  deg_count  <<<cdiv_ll(E, T), T, 0, stream>>>(ei, deg, E);
  compute_dis<<<cdiv_ll(n, T), T, 0, stream>>>(deg, dis, n);

  // layer 0 (one-hot + GEMM fused)
  embed_gemm0<<<cdiv_ll(nh, T), T, 0, stream>>>(x, W0, b0, h0, n);

  // conv1: h0 -> buf1 (relu)
  gemm_h_w_wmma<<<gemmGrid, T, 0, stream>>>(h0, W1, tmp, n);
  zero_f32     <<<cdiv_ll(nh, T), T, 0, stream>>>(buf1, nh);
  edge_scatter <<<cdiv_ll((long long)E * 16, T), T, 0, stream>>>(ei, dis, tmp, buf1, E);
  conv_finalize<<<cdiv_ll(nh, T), T, 0, stream>>>(buf1, tmp, dis, b1, nullptr, 1, n);

  // conv2: buf1 -> buf2 (relu)
  gemm_h_w_wmma<<<gemmGrid, T, 0, stream>>>(buf1, W2, tmp, n);
  zero_f32     <<<cdiv_ll(nh, T), T, 0, stream>>>(buf2, nh);
  edge_scatter <<<cdiv_ll((long long)E * 16, T), T, 0, stream>>>(ei, dis, tmp, buf2, E);
  conv_finalize<<<cdiv_ll(nh, T), T, 0, stream>>>(buf2, tmp, dis, b2, nullptr, 1, n);

  // conv3: buf2 -> buf1, + residual h0, relu
  gemm_h_w_wmma<<<gemmGrid, T, 0, stream>>>(buf2, W3, tmp, n);
  zero_f32     <<<cdiv_ll(nh, T), T, 0, stream>>>(buf1, nh);
  edge_scatter <<<cdiv_ll((long long)E * 16, T), T, 0, stream>>>(ei, dis, tmp, buf1, E);
  conv_finalize<<<cdiv_ll(nh, T), T, 0, stream>>>(buf1, tmp, dis, b3, h0, 1, n);

  // cluster assignments
  cluster_softmax<<<cdiv_ll(n, T), T, 0, stream>>>(buf1, Wl, bl, sm, n);

  // mincut + ortho losses
  zero_f32      <<<cdiv_ll(264, T), T, 0, stream>>>(ssm, 264);   // ss(256) + red(8)
  edge_num_reduce<<<cdiv_ll(E, T), T, 0, stream>>>(ei, sm, red, E);
  den_ss_reduce <<<cdiv_ll(n, T), T, 0, stream>>>(sm, deg, red, ssm, n);
  final_losses  <<<1, 256, 0, stream>>>(red, ssm, (float*)d_out);
}